// Convolutional_50474455662911
// MI455X (gfx1250) — compile-verified
//
#include <hip/hip_runtime.h>
#include <hip/hip_bf16.h>

typedef __attribute__((ext_vector_type(2))) float v2f;
typedef __attribute__((ext_vector_type(8))) float v8f;

#define IN_C   64
#define IMG_H  256
#define IMG_W  256
#define OUT_C  128
#define OH     254
#define OW     254
#define KTOT   576           // IN_C * 9
#define TILE_N 64            // output pixels per block (one row strip, 4 x 16 subtiles)
#define PATCH_W 66           // TILE_N + 2 halo
#define LDS_STRIDE 416       // floats per icPair: 416 % 64 == 32 -> half-waves on disjoint banks, 8B aligned
#define BLOCK  128           // 4 waves; wave w owns oc [32w, 32w+32)

// ---------------------------------------------------------------------------
// Repack weights [OUT_C][IN_C][3][3] -> W3[k4][half][oc][2] so each lane's
// A-fragment (V_WMMA_F32_16X16X4_F32 A layout) is one contiguous b64 load:
//   lane L (0..15):  v0=A[M=L][k4*4+0], v1=A[L][k4*4+1]   (half=0)
//   lane L+16:       v0=A[L][k4*4+2],   v1=A[L][k4*4+3]   (half=1)
// with k = (kh*3+kw)*64 + ic.
// ---------------------------------------------------------------------------
__global__ __launch_bounds__(256) void repack_weights_k(
    const float* __restrict__ w, float* __restrict__ w3) {
  int i = blockIdx.x * 256 + threadIdx.x;          // dst flat index, 73728 total
  if (i >= OUT_C * KTOT) return;
  int j    = i & 1;
  int t    = i >> 1;
  int oc   = t & (OUT_C - 1);
  t      >>= 7;
  int half = t & 1;
  int k4   = t >> 1;
  int k    = (k4 << 2) + (half << 1) + j;
  int khkw = k >> 6;                               // 0..8
  int ic   = k & 63;
  int kh   = khkw / 3;
  int kw   = khkw - kh * 3;
  w3[i] = w[((oc * IN_C + ic) * 3 + kh) * 3 + kw];
}

// ---------------------------------------------------------------------------
// Implicit-GEMM conv: one block = 128 oc x 64 pixels of one output row.
// 4 waves; wave w owns oc tile [32w,32w+32) with 8 v8f accumulators
// (2 oc subtiles x 4 px subtiles). Per K=4 step: 2 global b64 (A, L2-resident)
// + 4 ds b64 (B, 2addr-fusable) feed 8 WMMAs.
// ---------------------------------------------------------------------------
__global__ __launch_bounds__(BLOCK) void conv_wmma_f32_k(
    const float* __restrict__ x, const float* __restrict__ w3,
    const float* __restrict__ bias, float* __restrict__ out) {
  __shared__ float smem[32 * LDS_STRIDE];          // 52 KB: [icPair][ (r*66+c)*2 + parity ]

  const int owb = blockIdx.x * TILE_N;
  const int oh  = blockIdx.y;
  const int tid = threadIdx.x;

  // ---- stage input patch: rows [oh, oh+2], cols [owb, owb+65], all 64 ic ----
  for (int t = tid; t < IN_C * 3 * PATCH_W; t += BLOCK) {
    int ic  = t / (3 * PATCH_W);
    int rem = t - ic * (3 * PATCH_W);
    int r   = rem / PATCH_W;
    int c   = rem - r * PATCH_W;
    int gx  = owb + c;
    float v = 0.0f;
    if (gx < IMG_W) v = x[ic * (IMG_H * IMG_W) + (oh + r) * IMG_W + gx];
    smem[(ic >> 1) * LDS_STRIDE + (r * PATCH_W + c) * 2 + (ic & 1)] = v;
  }
  __syncthreads();

  const int wave   = tid >> 5;
  const int lane   = tid & 31;
  const int laneLo = lane & 15;
  const int hi     = lane >> 4;                    // 0: K{0,1}/M r ; 1: K{2,3}/M r+8
  const int ocb    = wave << 5;                    // 32 oc per wave

  v8f a0p0 = {}, a0p1 = {}, a0p2 = {}, a0p3 = {};  // oc subtile 0, px +0/+16/+32/+48
  v8f a1p0 = {}, a1p1 = {}, a1p2 = {}, a1p3 = {};  // oc subtile 1 (+16 oc)

  // A-fragment streams: advance 2*128*2 = 512 floats per K=4 step.
  const float* wp0 = w3 + ((hi * OUT_C) + (ocb + laneLo)) * 2;
  const float* wp1 = wp0 + 32;                     // +16 oc

  for (int kh = 0; kh < 3; ++kh) {
    for (int kw = 0; kw < 3; ++kw) {
      const int colbase = (kh * PATCH_W + laneLo + kw) * 2;
#pragma unroll
      for (int icq = 0; icq < 16; ++icq) {         // ic0 = icq*4
        v2f a0 = *(const v2f*)wp0;                 // global_load_b64
        v2f a1 = *(const v2f*)wp1;
        wp0 += 512;
        wp1 += 512;
        const int base = (icq * 2 + hi) * LDS_STRIDE + colbase;
        v2f b0 = *(const v2f*)&smem[base];         // k{+0,+1} / k{+2,+3}, px +0
        v2f b1 = *(const v2f*)&smem[base + 32];    // px +16
        v2f b2 = *(const v2f*)&smem[base + 64];    // px +32
        v2f b3 = *(const v2f*)&smem[base + 96];    // px +48
        a0p0 = __builtin_amdgcn_wmma_f32_16x16x4_f32(
                   false, a0, false, b0, (short)0, a0p0, false, false);
        a0p1 = __builtin_amdgcn_wmma_f32_16x16x4_f32(
                   false, a0, false, b1, (short)0, a0p1, false, false);
        a0p2 = __builtin_amdgcn_wmma_f32_16x16x4_f32(
                   false, a0, false, b2, (short)0, a0p2, false, false);
        a0p3 = __builtin_amdgcn_wmma_f32_16x16x4_f32(
                   false, a0, false, b3, (short)0, a0p3, false, false);
        a1p0 = __builtin_amdgcn_wmma_f32_16x16x4_f32(
                   false, a1, false, b0, (short)0, a1p0, false, false);
        a1p1 = __builtin_amdgcn_wmma_f32_16x16x4_f32(
                   false, a1, false, b1, (short)0, a1p1, false, false);
        a1p2 = __builtin_amdgcn_wmma_f32_16x16x4_f32(
                   false, a1, false, b2, (short)0, a1p2, false, false);
        a1p3 = __builtin_amdgcn_wmma_f32_16x16x4_f32(
                   false, a1, false, b3, (short)0, a1p3, false, false);
      }
    }
  }

  // ---- store: D layout VGPR r -> M = r (lanes 0-15) / r+8 (lanes 16-31) ----
  const int ow0 = owb + laneLo;
#pragma unroll
  for (int r = 0; r < 8; ++r) {
    int ocA   = ocb + (hi << 3) + r;               // oc subtile 0
    int ocB   = ocA + 16;                          // oc subtile 1
    float bvA = bias[ocA];
    float bvB = bias[ocB];
    float* rowA = out + (ocA * OH + oh) * OW;
    float* rowB = out + (ocB * OH + oh) * OW;
    if (ow0 < OW) {
      rowA[ow0]      = a0p0[r] + bvA;
      rowB[ow0]      = a1p0[r] + bvB;
    }
    if (ow0 + 16 < OW) {
      rowA[ow0 + 16] = a0p1[r] + bvA;
      rowB[ow0 + 16] = a1p1[r] + bvB;
    }
    if (ow0 + 32 < OW) {
      rowA[ow0 + 32] = a0p2[r] + bvA;
      rowB[ow0 + 32] = a1p2[r] + bvB;
    }
    if (ow0 + 48 < OW) {
      rowA[ow0 + 48] = a0p3[r] + bvA;
      rowB[ow0 + 48] = a1p3[r] + bvB;
    }
  }
}

extern "C" void kernel_launch(void* const* d_in, const int* in_sizes, int n_in,
                              void* d_out, int out_size, void* d_ws, size_t ws_size,
                              hipStream_t stream) {
  const float* x    = (const float*)d_in[0];
  const float* kern = (const float*)d_in[1];
  const float* bias = (const float*)d_in[2];
  float* out = (float*)d_out;
  float* w3  = (float*)d_ws;                        // needs 128*576*4 = 294912 B

  repack_weights_k<<<(OUT_C * KTOT) / 256, 256, 0, stream>>>(kern, w3);

  dim3 grid((OW + TILE_N - 1) / TILE_N, OH);        // 4 x 254 blocks
  conv_wmma_f32_k<<<grid, BLOCK, 0, stream>>>(x, w3, bias, out);
}